// HOIContactLoss_84963043049837
// MI455X (gfx1250) — compile-verified
//
#include <hip/hip_runtime.h>
#include <hip/hip_bf16.h>

// HOI contact (Chamfer) loss for MI455X / gfx1250.
//
// d2[m][n] = |x_m|^2 + |y_n|^2 - 2 x_m . y_n   (K=3 -> augmented K=4 GEMM)
//   A row m = [x0, x1, x2, |x|^2],  B col n = [-2y0, -2y1, -2y2, 1],
//   C[m][n] = |y_n|^2  (per-lane broadcast)  => one V_WMMA_F32_16X16X4_F32
// per 16x16 tile of the distance matrix (2048 FLOP/instr).
//
// Validity masking is folded into the B/C operands (invalid col -> d2 >= 2e10),
// and the reference's  where(valid, max(d2,0), BIG)  is exactly one
// v_med3_f32(d, 0, BIG) per element. All mins run in the u32 domain
// (values are in [0, BIG] so IEEE order == u32 order), enabling v_min_u32 /
// v_min3_u32 and plain ds_min_u32 / global_atomic_min_u32 with no float
// canonicalization overhead. B-tile loads are software-pipelined one tile
// ahead so s_wait_loadcnt overlaps ~30 VALU ops.

#define BATCH 16
#define P1    6890
#define P2    4000
#define P1PAD 6896            // 431 * 16
#define MTILES 431
#define NTILES 250            // P2 / 16 exactly
#define BIGF  1.0e10f
#define HUGEF 2.0e10f

typedef float v2f __attribute__((ext_vector_type(2)));
typedef float v8f __attribute__((ext_vector_type(8)));

// ---------------------------------------------------------------- prep ------
__global__ __launch_bounds__(256) void hoi_prep(
    const float* __restrict__ sv, const float* __restrict__ ov,
    float* __restrict__ xa4, float* __restrict__ yb4,
    unsigned* __restrict__ colmin_g, float* __restrict__ out)
{
    const int b = blockIdx.y;
    const int j = blockIdx.x * blockDim.x + threadIdx.x;
    if (b == 0 && j == 0) out[0] = 0.0f;                 // d_out is poisoned

    if (j < P1PAD) {
        // Pad rows with a far-away point so padded rows give huge d2 and can
        // never win a column-min.
        float x0 = 1.0e4f, x1 = 1.0e4f, x2 = 1.0e4f, n2 = 3.0e8f;
        if (j < P1) {
            const float* p = sv + ((size_t)b * P1 + j) * 3;
            x0 = p[0]; x1 = p[1]; x2 = p[2];
            n2 = x0 * x0 + x1 * x1 + x2 * x2;
        }
        ((float4*)xa4)[(size_t)b * P1PAD + j] = make_float4(x0, x1, x2, n2);
    }
    if (j < P2) {
        const float* p = ov + ((size_t)b * P2 + j) * 3;
        const float y0 = p[0], y1 = p[1], y2 = p[2];
        const float n2 = y0 * y0 + y1 * y1 + y2 * y2;
        ((float4*)yb4)[(size_t)b * P2 + j] =
            make_float4(-2.0f * y0, -2.0f * y1, -2.0f * y2, n2);
        colmin_g[(size_t)b * P2 + j] = __float_as_uint(BIGF);
    }
}

// ---------------------------------------------------------------- main ------
// Block = (batch, 8 waves x 16-row strips). Each wave owns 16 rows of x and
// sweeps all 250 column tiles with V_WMMA_F32_16X16X4_F32.
__global__ __launch_bounds__(256) void hoi_cham(
    const float* __restrict__ xa4f, const float* __restrict__ yb4f,
    const long long* __restrict__ vn,     // object_verts_n is int64 in the ref
    float* __restrict__ chamx, unsigned* __restrict__ colmin_g)
{
    __shared__ unsigned lcol[P2];                        // 16 KB of 320 KB LDS

    const int b   = blockIdx.y;
    const int tid = threadIdx.x;
    for (int i = tid; i < P2; i += 256) lcol[i] = __float_as_uint(BIGF);
    __syncthreads();

    const int  wave   = tid >> 5;
    const int  lane   = tid & 31;
    const int  lh     = lane & 15;
    const bool hiHalf = lane >= 16;
    const int  mstrip = blockIdx.x * 8 + wave;
    const int  nvalid = (int)vn[b];

    if (mstrip < MTILES) {                               // wave-uniform branch
        const float4* xa4 = (const float4*)xa4f;
        const float4* ybp = (const float4*)yb4f + (size_t)b * P2 + lh;
        const int m_base = mstrip * 16;

        // A matrix (16x4 f32, 2 VGPRs): lane = M; VGPR0 holds K0|K2, VGPR1 K1|K3.
        const float4 av = xa4[(size_t)b * P1PAD + m_base + lh];
        v2f A;
        A[0] = hiHalf ? av.z : av.x;                     // K=0 | K=2
        A[1] = hiHalf ? av.w : av.y;                     // K=1 | K=3 (=|x|^2)

        unsigned rowmin[8];
#pragma unroll
        for (int i = 0; i < 8; ++i) rowmin[i] = __float_as_uint(BIGF);

        float4 bv_next = ybp[0];                         // pipeline: tile 0
        for (int nt = 0; nt < NTILES; ++nt) {
            const float4 bv = bv_next;
            // Prefetch next tile (last iter reads into our own chamx region:
            // allocated workspace, value unused).
            bv_next = ybp[(nt + 1) * 16];

            const int  n     = nt * 16 + lh;
            const bool valid = n < nvalid;

            // B matrix (4x16, 2 VGPRs), mirror of A layout: lane = N.
            // Fold the validity mask into the operands: invalid columns get
            // B=[0,0,0,1], C=2e10  =>  D = |x|^2 + 2e10 >= 2e10.
            v2f Bm;
            Bm[0] = valid ? (hiHalf ? bv.z : bv.x) : 0.0f;   // K=0 | K=2 (-2y)
            Bm[1] = hiHalf ? 1.0f : (valid ? bv.y : 0.0f);   // K=1 | K=3 (=1)
            const float cw = valid ? bv.w : HUGEF;           // |y_n|^2 | huge
            v8f C;
#pragma unroll
            for (int i = 0; i < 8; ++i) C[i] = cw;

            // D[m][n] = -2 x.y + |x|^2 + |y|^2   (or >= 2e10 if masked)
            v8f D = __builtin_amdgcn_wmma_f32_16x16x4_f32(
                false, A, false, Bm, (short)0, C, false, false);

            unsigned cu = __float_as_uint(BIGF);
#pragma unroll
            for (int i = 0; i < 8; ++i) {
                // where(valid, max(d2,0), BIG) == med3(d, 0, BIG): one VALU op.
                const unsigned du =
                    __float_as_uint(__builtin_amdgcn_fmed3f(D[i], 0.0f, BIGF));
                rowmin[i] = du < rowmin[i] ? du : rowmin[i]; // v_min_u32
                cu        = du < cu        ? du : cu;        // v_min3_u32 tree
            }
            // Column-min of this tile (rows split across the two half-waves);
            // both lanes L and L+16 hit column n -> single ds_min_u32.
            atomicMin(&lcol[n], cu);
        }

        // Row-min: butterfly within each 16-lane half (u32 order == f32 order).
#pragma unroll
        for (int i = 0; i < 8; ++i) {
            unsigned r = rowmin[i], t;
            t = __shfl_xor(r, 1, 32); r = t < r ? t : r;
            t = __shfl_xor(r, 2, 32); r = t < r ? t : r;
            t = __shfl_xor(r, 4, 32); r = t < r ? t : r;
            t = __shfl_xor(r, 8, 32); r = t < r ? t : r;
            rowmin[i] = r;
        }
        if (lh == 0) {                                   // lanes 0 and 16
            const int mrow = m_base + (hiHalf ? 8 : 0);  // VGPR i -> row i / i+8
            float4* cx = (float4*)(chamx + (size_t)b * P1PAD + mrow);
            cx[0] = make_float4(__uint_as_float(rowmin[0]), __uint_as_float(rowmin[1]),
                                __uint_as_float(rowmin[2]), __uint_as_float(rowmin[3]));
            cx[1] = make_float4(__uint_as_float(rowmin[4]), __uint_as_float(rowmin[5]),
                                __uint_as_float(rowmin[6]), __uint_as_float(rowmin[7]));
        }
    }

    __syncthreads();
    for (int i = tid; i < P2; i += 256)
        atomicMin(&colmin_g[(size_t)b * P2 + i], lcol[i]);
}

// ------------------------------------------------------------- finalize -----
__global__ __launch_bounds__(256) void hoi_finalize(
    const float* __restrict__ smap, const float* __restrict__ omap,
    const long long* __restrict__ vn,
    const float* __restrict__ chamx, const unsigned* __restrict__ colmin_g,
    float* __restrict__ out)
{
    __shared__ float s0[256], s1[256], s2[256], s3[256];
    const int b = blockIdx.x, tid = threadIdx.x;
    const int nvalid = (int)vn[b];

    float sxn = 0.f, sxd = 0.f, syn = 0.f, syd = 0.f;
    for (int m = tid; m < P1; m += 256) {
        const float w = smap[(size_t)b * P1 + m];
        sxn += w * chamx[(size_t)b * P1PAD + m];
        sxd += w;
    }
    for (int n = tid; n < P2; n += 256) {
        if (n < nvalid) {
            const float w = omap[(size_t)b * P2 + n];
            syn += w * __uint_as_float(colmin_g[(size_t)b * P2 + n]);
            syd += w;
        }
    }
    s0[tid] = sxn; s1[tid] = sxd; s2[tid] = syn; s3[tid] = syd;
    __syncthreads();
    for (int off = 128; off > 0; off >>= 1) {
        if (tid < off) {
            s0[tid] += s0[tid + off]; s1[tid] += s1[tid + off];
            s2[tid] += s2[tid + off]; s3[tid] += s3[tid + off];
        }
        __syncthreads();
    }
    if (tid == 0) {
        const float loss = s0[0] / (s1[0] + 1e-6f) + s2[0] / (s3[0] + 1e-6f);
        atomicAdd(out, loss * (1.0f / (float)BATCH));
    }
}

// --------------------------------------------------------------- launch -----
extern "C" void kernel_launch(void* const* d_in, const int* in_sizes, int n_in,
                              void* d_out, int out_size, void* d_ws, size_t ws_size,
                              hipStream_t stream)
{
    const float*     sv   = (const float*)d_in[0];       // smpl_v        f32
    const float*     ov   = (const float*)d_in[1];       // object_v      f32
    const float*     smap = (const float*)d_in[2];       // smpl maps     f32
    const float*     omap = (const float*)d_in[3];       // object maps   f32
    const long long* vn   = (const long long*)d_in[4];   // verts_n       i64
    float* out = (float*)d_out;

    // Workspace layout (floats), ~3.5 MB total.  NOTE: yb4 must be followed by
    // more workspace (chamx) because the pipelined B load reads one tile past
    // the end of yb4 for the last batch.
    float*    ws      = (float*)d_ws;
    float*    xa4     = ws;                                  // B*P1PAD*4
    float*    yb4     = xa4 + (size_t)BATCH * P1PAD * 4;     // B*P2*4
    float*    chamx   = yb4 + (size_t)BATCH * P2 * 4;        // B*P1PAD
    unsigned* colming = (unsigned*)(chamx + (size_t)BATCH * P1PAD); // B*P2

    dim3 gp((P1PAD + 255) / 256, BATCH);
    hoi_prep<<<gp, 256, 0, stream>>>(sv, ov, xa4, yb4, colming, out);

    dim3 gm((MTILES + 7) / 8, BATCH);                    // 54 x 16 blocks
    hoi_cham<<<gm, 256, 0, stream>>>(xa4, yb4, vn, chamx, colming);

    hoi_finalize<<<BATCH, 256, 0, stream>>>(smap, omap, vn, chamx, colming, out);
}